// QCPLayer_53781580480687
// MI455X (gfx1250) — compile-verified
//
#include <hip/hip_runtime.h>
#include <math.h>

#define N_QUBITS 12
#define N_LAYERS 4
#define BATCH    256
#define DIM      4096   // 2^12 amplitudes per sample

typedef __attribute__((ext_vector_type(2))) float v2f;
typedef __attribute__((ext_vector_type(8))) float v8f;

// D(16x16,f32) += A(16x4,f32) x B(4x16,f32)  -- CDNA5 FP32 WMMA
__device__ __forceinline__ v8f wmma_f32(v2f a, v2f b, v8f c) {
  // (neg_a, A, neg_b, B, c_mod, C, reuse_a, reuse_b); F32 has no A/B neg -> false
  return __builtin_amdgcn_wmma_f32_16x16x4_f32(false, a, false, b, (short)0, c,
                                               false, false);
}

// In-group CX cascade permutation on 4-bit group index (b3 = MSB = first qubit
// of group): CX(q0,q0+1);CX(q0+1,q0+2);CX(q0+2,q0+3)  =>  b2^=b3; b1^=b2; b0^=b1
__device__ __forceinline__ int cx_perm4(int i) {
  int i3 = (i >> 3) & 1, i2 = (i >> 2) & 1, i1 = (i >> 1) & 1, i0 = i & 1;
  int n2 = i2 ^ i3, n1 = i1 ^ n2, n0 = i0 ^ n1;
  return (i3 << 3) | (n2 << 2) | (n1 << 1) | n0;
}

// ---------------------------------------------------------------------------
// Kernel 0: build per-layer fused 16x16 complex group matrices into d_ws.
// Slots per layer: 0=M_A, 1=M_B0, 2=M_B1, 3=M_C0, 4=M_C1.
// Storage per matrix: Re[16][16] then Im[16][16] (512 floats, row-major).
// M_A  = P * (R0 (x) R1 (x) R2 (x) R3)
// M_B0 = P * T_B          M_B1 = P * X_msb * T_B   (selected by bit3 of state)
// M_C0 = P * T_C          M_C1 = P * X_msb * T_C   (selected by bit7 of state)
// ---------------------------------------------------------------------------
__global__ void qc_build_mats(const float* __restrict__ w,
                              float* __restrict__ M) {
  const int t = threadIdx.x;   // 0..255
  const int i = t >> 4;        // pre-permutation row
  const int j = t & 15;        // column
  for (int l = 0; l < N_LAYERS; ++l) {
    for (int g = 0; g < 3; ++g) {
      const int q0 = g * 4;
      float Tr = 1.f, Ti = 0.f;
      #pragma unroll
      for (int tq = 0; tq < 4; ++tq) {
        const int q = q0 + tq;
        const float a = w[(l * 12 + q) * 2 + 0];   // RY angle
        const float b = w[(l * 12 + q) * 2 + 1];   // RZ angle
        float sa, ca, sb, cb;
        __sincosf(0.5f * a, &sa, &ca);
        __sincosf(0.5f * b, &sb, &cb);
        const int bi = (i >> (3 - tq)) & 1;
        const int bj = (j >> (3 - tq)) & 1;
        // R = RZ(b)*RY(a): row0 *= e^{-ib/2}, row1 *= e^{+ib/2}
        float rr, ri;
        if (bi == 0) {
          const float mm = (bj == 0) ? ca : -sa;
          rr = cb * mm; ri = -sb * mm;
        } else {
          const float mm = (bj == 0) ? sa : ca;
          rr = cb * mm; ri = sb * mm;
        }
        const float nr = Tr * rr - Ti * ri;
        const float ni = Tr * ri + Ti * rr;
        Tr = nr; Ti = ni;
      }
      const int pi  = cx_perm4(i);       // row after CX cascade
      const int pif = cx_perm4(i ^ 8);   // row after X_msb then CX cascade
      if (g == 0) {
        const int base = (l * 5 + 0) * 512;
        M[base + pi * 16 + j]       = Tr;
        M[base + 256 + pi * 16 + j] = Ti;
      } else {
        const int s0 = (g == 1) ? 1 : 3;
        const int b0 = (l * 5 + s0) * 512;
        const int b1 = (l * 5 + s0 + 1) * 512;
        M[b0 + pi * 16 + j]        = Tr;  M[b0 + 256 + pi * 16 + j]  = Ti;
        M[b1 + pif * 16 + j]       = Tr;  M[b1 + 256 + pif * 16 + j] = Ti;
      }
    }
  }
}

// ---------------------------------------------------------------------------
// Kernel 1: one workgroup (8 wave32) per batch sample; state in LDS.
// Basis index s = sum_q bit_q * 2^(11-q)  (qubit 0 = MSB, matching jax layout).
// Per layer: diag RZ(x) phase; GEMM pass on group {0-3}; group {4-7} with
// matrix chosen by bit3; group {8-11} with matrix chosen by bit7.
// ---------------------------------------------------------------------------
__global__ void __launch_bounds__(256) qc_sim(const float* __restrict__ x,
                                              const float* __restrict__ Mg,
                                              float* __restrict__ out) {
  __shared__ float sRe[DIM];
  __shared__ float sIm[DIM];
  __shared__ float sMat[5 * 512];
  __shared__ float sHX[N_QUBITS];
  __shared__ float sRed[256];

  const int tid  = threadIdx.x;
  const int smp  = blockIdx.x;
  const int lane = tid & 31;
  const int wave = tid >> 5;
  const int m    = lane & 15;   // A-row / B-col / D-col for this lane
  const int hi   = lane >> 4;   // lane half selects K pair / D row half

  if (tid < N_QUBITS) sHX[tid] = 0.5f * x[smp * N_QUBITS + tid];

  // |psi> = H^{(x)12} |0> = uniform real amplitude 2^-6
  for (int i = 0; i < 16; ++i) {
    sRe[tid + 256 * i] = 0.015625f;
    sIm[tid + 256 * i] = 0.f;
  }
  __syncthreads();

  for (int l = 0; l < N_LAYERS; ++l) {
    // stage this layer's 5 complex 16x16 matrices into LDS (L2-broadcast read)
    const float* src = Mg + l * 5 * 512;
    for (int i = tid; i < 5 * 512; i += 256) sMat[i] = src[i];

    // ---- diagonal per-sample RZ(x) pass: amp *= e^{i*phi(s)} ----
    for (int i = 0; i < 16; ++i) {
      const int s = tid + 256 * i;
      float ang = 0.f;
      #pragma unroll
      for (int q = 0; q < N_QUBITS; ++q)
        ang += ((s >> (11 - q)) & 1) ? sHX[q] : -sHX[q];
      float sn, cs;
      __sincosf(ang, &sn, &cs);
      const float r = sRe[s], im = sIm[s];
      sRe[s] = r * cs - im * sn;
      sIm[s] = r * sn + im * cs;
    }
    __syncthreads();  // also covers sMat staging

    // ---- Pass A: group {0-3}.  State as S[g0][col] (row stride 256).
    //      O = M_A * S, 16 column tiles of 16; wave-uniform loops keep EXEC=~0.
    for (int tile = wave; tile < 16; tile += 8) {
      const int cb = tile * 16 + m;          // my column
      v8f accR = {}; v8f accI = {};
      #pragma unroll
      for (int kc = 0; kc < 4; ++kc) {
        const int kb = kc * 4 + hi * 2;
        const float ar0 = sMat[m * 16 + kb],       ar1 = sMat[m * 16 + kb + 1];
        const float ai0 = sMat[256 + m * 16 + kb], ai1 = sMat[256 + m * 16 + kb + 1];
        const float br0 = sRe[kb * 256 + cb],      br1 = sRe[(kb + 1) * 256 + cb];
        const float bi0 = sIm[kb * 256 + cb],      bi1 = sIm[(kb + 1) * 256 + cb];
        const v2f aR = {ar0, ar1}, aI = {ai0, ai1}, aN = {-ai0, -ai1};
        const v2f bR = {br0, br1}, bI = {bi0, bi1};
        accR = wmma_f32(aR, bR, accR);   // Re += Ur*Sr
        accR = wmma_f32(aN, bI, accR);   // Re -= Ui*Si
        accI = wmma_f32(aR, bI, accI);   // Im += Ur*Si
        accI = wmma_f32(aI, bR, accI);   // Im += Ui*Sr
      }
      #pragma unroll
      for (int r = 0; r < 8; ++r) {
        const int row = r + hi * 8;
        sRe[row * 256 + cb] = accR[r];
        sIm[row * 256 + cb] = accI[r];
      }
    }
    __syncthreads();

    // ---- Pass B: group {4-7}.  For each h (top 4 bits): O_h = M_B{b3} * S_h,
    //      S_h contiguous 16x16 block at h*256, row stride 16.
    for (int tile = wave; tile < 16; tile += 8) {
      const int h    = tile;
      const int base = h * 256;
      const int mb   = (1 + (h & 1)) * 512;   // bit3 of state selects matrix
      v8f accR = {}; v8f accI = {};
      #pragma unroll
      for (int kc = 0; kc < 4; ++kc) {
        const int kb = kc * 4 + hi * 2;
        const float ar0 = sMat[mb + m * 16 + kb],       ar1 = sMat[mb + m * 16 + kb + 1];
        const float ai0 = sMat[mb + 256 + m * 16 + kb], ai1 = sMat[mb + 256 + m * 16 + kb + 1];
        const float br0 = sRe[base + kb * 16 + m],      br1 = sRe[base + (kb + 1) * 16 + m];
        const float bi0 = sIm[base + kb * 16 + m],      bi1 = sIm[base + (kb + 1) * 16 + m];
        const v2f aR = {ar0, ar1}, aI = {ai0, ai1}, aN = {-ai0, -ai1};
        const v2f bR = {br0, br1}, bI = {bi0, bi1};
        accR = wmma_f32(aR, bR, accR);
        accR = wmma_f32(aN, bI, accR);
        accI = wmma_f32(aR, bI, accI);
        accI = wmma_f32(aI, bR, accI);
      }
      #pragma unroll
      for (int r = 0; r < 8; ++r) {
        const int row = r + hi * 8;
        sRe[base + row * 16 + m] = accR[r];
        sIm[base + row * 16 + m] = accI[r];
      }
    }
    __syncthreads();

    // ---- Pass C: group {8-11}.  O[u][i] = sum_j S[u][j] * M_C{b7}[i][j],
    //      i.e. O = S * M^T.  b7 = u&1 varies, so tile rows share parity p:
    //      u = ut*32 + 2*m + p  -> one matrix per tile.
    for (int tile = wave; tile < 16; tile += 8) {
      const int ut = tile >> 1, p = tile & 1;
      const int mc = (3 + p) * 512;
      v8f accR = {}; v8f accI = {};
      #pragma unroll
      for (int kc = 0; kc < 4; ++kc) {
        const int kb = kc * 4 + hi * 2;
        const int ua = (ut * 32 + 2 * m + p) * 16;      // my state row
        const float ar0 = sRe[ua + kb], ar1 = sRe[ua + kb + 1];
        const float ai0 = sIm[ua + kb], ai1 = sIm[ua + kb + 1];
        const float br0 = sMat[mc + m * 16 + kb],       br1 = sMat[mc + m * 16 + kb + 1];
        const float bi0 = sMat[mc + 256 + m * 16 + kb], bi1 = sMat[mc + 256 + m * 16 + kb + 1];
        const v2f aR = {ar0, ar1}, aI = {ai0, ai1}, aN = {-ai0, -ai1};
        const v2f bR = {br0, br1}, bI = {bi0, bi1};
        accR = wmma_f32(aR, bR, accR);   // Re += Sr*Mr^T
        accR = wmma_f32(aN, bI, accR);   // Re -= Si*Mi^T
        accI = wmma_f32(aR, bI, accI);   // Im += Sr*Mi^T
        accI = wmma_f32(aI, bR, accI);   // Im += Si*Mr^T
      }
      #pragma unroll
      for (int r = 0; r < 8; ++r) {
        const int uo = (ut * 32 + 2 * (r + hi * 8) + p) * 16;
        sRe[uo + m] = accR[r];
        sIm[uo + m] = accI[r];
      }
    }
    __syncthreads();
  }

  // ---- <Z(0)> = P(bit0=0) - P(bit0=1); bit0 = s>>11 ----
  float part = 0.f;
  for (int i = 0; i < 16; ++i) {
    const int s = tid + 256 * i;
    const float r = sRe[s], im = sIm[s];
    const float p2 = r * r + im * im;
    part += (i < 8) ? p2 : -p2;     // s >= 2048  <=>  i >= 8
  }
  sRed[tid] = part;
  __syncthreads();
  for (int off = 128; off > 0; off >>= 1) {
    if (tid < off) sRed[tid] += sRed[tid + off];
    __syncthreads();
  }
  if (tid == 0) out[smp] = sRed[0];
}

// ---------------------------------------------------------------------------
extern "C" void kernel_launch(void* const* d_in, const int* in_sizes, int n_in,
                              void* d_out, int out_size, void* d_ws,
                              size_t ws_size, hipStream_t stream) {
  (void)in_sizes; (void)n_in; (void)out_size; (void)ws_size;
  const float* x = (const float*)d_in[0];   // (256, 12) float32
  const float* w = (const float*)d_in[1];   // (96,)     float32
  float* mats = (float*)d_ws;               // 4*5*512 floats = 40 KB
  qc_build_mats<<<1, 256, 0, stream>>>(w, mats);
  qc_sim<<<BATCH, 256, 0, stream>>>(x, mats, (float*)d_out);
}